// VarifoldKernel_28690381537757
// MI455X (gfx1250) — compile-verified
//
#include <hip/hip_runtime.h>

typedef __attribute__((ext_vector_type(2))) float v2f;
typedef __attribute__((ext_vector_type(8))) float v8f;

#define KPAD 68      // 3 pos dims *scaled* + 1 zero pad + 64 feat dims
#define KSTEPS 17    // 68 / 4 per V_WMMA_F32_16X16X4_F32
#define TILE_FLOATS (16 * KPAD)      // 1088 floats = 4352 B per B tile
#define TILE_CHUNKS (TILE_FLOATS / 4)  // 272 float4 chunks

// ---------------------------------------------------------------------------
// Pack [2*pos | 0 | featScale*feat] rows (KPAD floats) and per-point log-weight
//   lw = log(w) - 2*|pos|^2
// ---------------------------------------------------------------------------
__global__ void varifold_prep(const float* __restrict__ pos,
                              const float* __restrict__ feat,
                              const float* __restrict__ w,
                              float* __restrict__ outMat,
                              float* __restrict__ outLw,
                              int total, float featScale) {
    int idx = blockIdx.x * blockDim.x + threadIdx.x;
    if (idx >= total) return;
    const float* p = pos + (size_t)idx * 3;
    const float* f = feat + (size_t)idx * 64;
    float* o = outMat + (size_t)idx * KPAD;
    float sq = 0.0f;
#pragma unroll
    for (int d = 0; d < 3; ++d) {
        float v = p[d];
        sq += v * v;
        o[d] = 2.0f * v;           // 2*pos (A) x 2*pos (B) -> 4*cross
    }
    o[3] = 0.0f;                   // pad K=3 -> K=4
#pragma unroll
    for (int j = 0; j < 64; ++j)
        o[4 + j] = featScale * f[j];   // A: +1.0 * feat1, B: -0.5 * feat2
    outLw[idx] = __logf(w[idx]) - 2.0f * sq;
}

// ---------------------------------------------------------------------------
// Main: 256-thread block = 8 wave32, block covers 128 rows (wave w owns row
// tile blockIdx.x*8 + w) and sweeps ctPerBlock 16-column tiles. Each B tile
// (16 rows x KPAD floats, contiguous 4352 B) is staged once per block into
// double-buffered LDS by a cooperative float4 copy; all 8 waves then read
// their WMMA fragments via ds_load (conflict-free), giving 8x L2-traffic
// reduction vs per-wave global reads. Next tile's global loads are issued
// before the current tile's 17-WMMA chain + exp epilogue so their wait (at
// the ds_store) is fully covered by matrix math.
// ---------------------------------------------------------------------------
__global__ __launch_bounds__(256) void varifold_main(
        const float* __restrict__ Amat, const float* __restrict__ Bmat,
        const float* __restrict__ lw1, const float* __restrict__ lw2,
        float* __restrict__ partial, int N, int M, int ctPerBlock) {
    const int b = blockIdx.z;
    const int rowTile = blockIdx.x * 8 + (threadIdx.x >> 5);
    const int lane = threadIdx.x & 31;
    const int wave = threadIdx.x >> 5;
    const int tid = threadIdx.x;
    const int hi = lane >> 4;      // 0: K={0,1} half, 1: K={2,3} half
    const int lo = lane & 15;      // row (A) / col (B,C)

    __shared__ float bbuf[2][TILE_FLOATS];

    // Per-lane A fragments: A'[b][rowTile*16 + lo][2*hi + 4*s]
    const float* aptr = Amat + ((size_t)b * N + (size_t)rowTile * 16 + lo) * KPAD + 2 * hi;
    v2f afrag[KSTEPS];
#pragma unroll
    for (int s = 0; s < KSTEPS; ++s)
        afrag[s] = *(const v2f*)(aptr + 4 * s);

    // Row log-weights for this lane's C-layout rows: M_local = r + 8*hi
    const float* lw1p = lw1 + (size_t)b * N + (size_t)rowTile * 16 + 8 * hi;
    float lw1v[8];
#pragma unroll
    for (int r = 0; r < 8; ++r) lw1v[r] = lw1p[r];

    const int ct0 = blockIdx.y * ctPerBlock;     // first column tile
    const float* Bbase = Bmat + (size_t)b * M * KPAD;
    const float* lw2b  = lw2 + (size_t)b * M;

    // Staging helpers: chunk c0 always valid (256 < 272), c1 for tid < 16.
    const int c1 = tid + 256;
    float4* dst0 = (float4*)bbuf[0];
    float4* dst1 = (float4*)bbuf[1];

    // Prologue: stage tile ct0 into buffer 0.
    {
        const float4* gsrc = (const float4*)(Bbase + (size_t)ct0 * 16 * KPAD);
        float4 p0 = gsrc[tid];
        float4 p1v;
        if (tid < TILE_CHUNKS - 256) p1v = gsrc[c1];
        dst0[tid] = p0;
        if (tid < TILE_CHUNKS - 256) dst0[c1] = p1v;
    }
    __syncthreads();

    // Per-lane fragment base inside an LDS tile.
    const int fragOff = lo * KPAD + 2 * hi;

    float sum = 0.0f;
    for (int i = 0; i < ctPerBlock; ++i) {
        const int ct = ct0 + i;
        const int bufSel = i & 1;
        const float* lbase = bbuf[bufSel] + fragOff;
        float4* dstn = bufSel ? dst0 : dst1;

        // 1. Prefetch next tile (clamped; last iteration re-reads, cached).
        const int ctn = (i + 1 < ctPerBlock) ? ct + 1 : ct;
        const float4* gsrc = (const float4*)(Bbase + (size_t)ctn * 16 * KPAD);
        float4 p0 = gsrc[tid];
        float4 p1v;
        if (tid < TILE_CHUNKS - 256) p1v = gsrc[c1];

        // 2. Fragments from LDS + dual-chain WMMA.
        v8f acc0 = {0.f, 0.f, 0.f, 0.f, 0.f, 0.f, 0.f, 0.f};
        v8f acc1 = {0.f, 0.f, 0.f, 0.f, 0.f, 0.f, 0.f, 0.f};
#pragma unroll
        for (int s = 0; s + 1 < KSTEPS; s += 2) {
            v2f bf0 = *(const v2f*)(lbase + 4 * s);
            v2f bf1 = *(const v2f*)(lbase + 4 * (s + 1));
            acc0 = __builtin_amdgcn_wmma_f32_16x16x4_f32(
                    false, afrag[s],     false, bf0, (short)0, acc0, false, false);
            acc1 = __builtin_amdgcn_wmma_f32_16x16x4_f32(
                    false, afrag[s + 1], false, bf1, (short)0, acc1, false, false);
        }
        {
            v2f bfl = *(const v2f*)(lbase + 4 * (KSTEPS - 1));
            acc0 = __builtin_amdgcn_wmma_f32_16x16x4_f32(
                    false, afrag[KSTEPS - 1], false, bfl, (short)0, acc0, false, false);
        }

        // Epilogue: exp(C + lw1[row] + lw2[col]) summed per lane.
        const float lw2v = lw2b[ct * 16 + lo];
#pragma unroll
        for (int r = 0; r < 8; ++r)
            sum += __expf(acc0[r] + acc1[r] + lw1v[r] + lw2v);

        // 3. Commit prefetched tile to the other buffer.
        dstn[tid] = p0;
        if (tid < TILE_CHUNKS - 256) dstn[c1] = p1v;

        // 4. One barrier per tile keeps the double buffer race-free.
        __syncthreads();
    }

    // wave32 reduction
#pragma unroll
    for (int off = 16; off >= 1; off >>= 1)
        sum += __shfl_down(sum, off, 32);

    __shared__ float red[8];
    if (lane == 0) red[wave] = sum;
    __syncthreads();
    if (tid == 0) {
        float t = 0.0f;
#pragma unroll
        for (int w = 0; w < 8; ++w) t += red[w];
        partial[((size_t)b * gridDim.y + blockIdx.y) * gridDim.x + blockIdx.x] = t;
    }
}

// ---------------------------------------------------------------------------
// Deterministic final reduction: one block per batch sums its partials.
// ---------------------------------------------------------------------------
__global__ __launch_bounds__(256) void varifold_reduce(
        const float* __restrict__ partial, float* __restrict__ out, int nParts) {
    const int b = blockIdx.x;
    const int tid = threadIdx.x;
    float sum = 0.0f;
    for (int i = tid; i < nParts; i += blockDim.x)
        sum += partial[(size_t)b * nParts + i];
#pragma unroll
    for (int off = 16; off >= 1; off >>= 1)
        sum += __shfl_down(sum, off, 32);
    __shared__ float red[8];
    if ((tid & 31) == 0) red[tid >> 5] = sum;
    __syncthreads();
    if (tid == 0) {
        float t = 0.0f;
#pragma unroll
        for (int w = 0; w < 8; ++w) t += red[w];
        out[b] = t;
    }
}

// ---------------------------------------------------------------------------
extern "C" void kernel_launch(void* const* d_in, const int* in_sizes, int n_in,
                              void* d_out, int out_size, void* d_ws, size_t ws_size,
                              hipStream_t stream) {
    const float* pos1  = (const float*)d_in[0];
    const float* feat1 = (const float*)d_in[1];
    const float* w1    = (const float*)d_in[2];
    const float* pos2  = (const float*)d_in[3];
    const float* feat2 = (const float*)d_in[4];
    const float* w2    = (const float*)d_in[5];
    float* out = (float*)d_out;

    const int B = out_size;                 // one scalar per batch
    const int N = in_sizes[2] / B;          // w1 has B*N elements
    const int M = in_sizes[5] / B;          // w2 has B*M elements

    // Workspace layout (floats)
    float* ws      = (float*)d_ws;
    float* Amat    = ws;
    float* Bmat    = Amat + (size_t)B * N * KPAD;
    float* lw1     = Bmat + (size_t)B * M * KPAD;
    float* lw2     = lw1 + (size_t)B * N;
    float* partial = lw2 + (size_t)B * M;

    const int t1 = B * N;
    varifold_prep<<<(t1 + 255) / 256, 256, 0, stream>>>(pos1, feat1, w1, Amat, lw1, t1, 1.0f);
    const int t2 = B * M;
    varifold_prep<<<(t2 + 255) / 256, 256, 0, stream>>>(pos2, feat2, w2, Bmat, lw2, t2, -0.5f);

    const int rowBlocks = N / 128;          // 8 row tiles per block
    const int mTiles = M / 16;
    int colBlocks = 4;                      // split columns for occupancy
    while (colBlocks > 1 && (mTiles % colBlocks)) colBlocks >>= 1;
    const int ctPerBlock = mTiles / colBlocks;

    dim3 grid(rowBlocks, colBlocks, B);
    varifold_main<<<grid, 256, 0, stream>>>(Amat, Bmat, lw1, lw2, partial, N, M, ctPerBlock);

    varifold_reduce<<<B, 256, 0, stream>>>(partial, out, rowBlocks * colBlocks);
}